// AdaFace_206158430559
// MI455X (gfx1250) — compile-verified
//
#include <hip/hip_runtime.h>
#include <hip/hip_bf16.h>
#include <math.h>

typedef __attribute__((ext_vector_type(2))) float v2f;
typedef __attribute__((ext_vector_type(8))) float v8f;

#define EPS_C   0.001f
#define M_C     0.4f
#define H_C     0.333f
#define S_C     64.0f
#define PI_C    3.14159265358979323846f

#define NROWS   512      // N
#define KDIM    512      // EMB
#define NCOLS   100000   // NUM_CLASSES

#define BM      128
#define BN      160
#define BNP     176      // padded pair-row stride: 2*BNP % 64(banks) == 32 -> conflict-free frags
#define BK      32

// ---------------------------------------------------------------------------
// Kernel 1: per-row margin parameters from norms (mean / unbiased std).
// ---------------------------------------------------------------------------
__global__ void adaface_margin_prep(const float* __restrict__ norms,
                                    float* __restrict__ gang,
                                    float* __restrict__ gadd) {
    __shared__ float s1[NROWS];
    __shared__ float s2[NROWS];
    const int i = threadIdx.x;
    float x = fminf(fmaxf(norms[i], 0.001f), 100.0f);
    s1[i] = x;
    s2[i] = x * x;
    __syncthreads();
    #pragma unroll
    for (int off = NROWS / 2; off > 0; off >>= 1) {
        if (i < off) { s1[i] += s1[i + off]; s2[i] += s2[i + off]; }
        __syncthreads();
    }
    const float mean = s1[0] * (1.0f / (float)NROWS);
    const float var  = (s2[0] - (float)NROWS * mean * mean) * (1.0f / (float)(NROWS - 1));
    const float sd   = sqrtf(fmaxf(var, 0.0f));
    float ms = (x - mean) / (sd + EPS_C) * H_C;
    ms = fminf(fmaxf(ms, -1.0f), 1.0f);
    gang[i] = -M_C * ms;
    gadd[i] = M_C + M_C * ms;
}

// ---------------------------------------------------------------------------
// Kernel 2: fused GEMM + AdaFace epilogue via V_WMMA_F32_16X16X4_F32.
// LDS B tile stored K-pair-interleaved: element (k,c) at ((k/2)*BNP + c)*2 + (k&1),
// so every B fragment is a single aligned ds_load_b64 into an even VGPR pair.
// Grid: (NCOLS/BN, NROWS/BM), block 256 (8 waves).
// ---------------------------------------------------------------------------
__global__ __launch_bounds__(256)
void adaface_gemm(const float* __restrict__ A,      // embeddings (N, K) row-major
                  const float* __restrict__ B,      // kernel (K, C) row-major
                  const int*   __restrict__ label,  // (N,)
                  const float* __restrict__ gang,   // (N,)
                  const float* __restrict__ gadd,   // (N,)
                  float*       __restrict__ out) {  // (N, C)
    __shared__ float Bs[(BK / 2) * BNP * 2];   // 16 pair-rows * 176 cols * float2 = 22 KB
    __shared__ float colinv[BN];

    const int tid  = threadIdx.x;          // 0..255
    const int wave = tid >> 5;             // 0..7 -> 16-row strip
    const int lane = tid & 31;
    const int lh   = lane >> 4;            // lane half (WMMA K/M select)
    const int ll   = lane & 15;

    const int c0 = blockIdx.x * BN;
    const int m0 = blockIdx.y * BM + wave * 16;

    v8f acc[10] = {};                      // 10 column tiles of 16x16 f32
    float csq = 0.0f;                      // per-column sum of squares (tid < BN)

    const float* __restrict__ Arow = A + (size_t)(m0 + ll) * KDIM;

    for (int kc = 0; kc < KDIM; kc += BK) {
        // 2x2-block cooperative load: 16 pair-rows x 80 col-pairs = 1280 blocks
        float2 lo[5], hi[5];
        #pragma unroll
        for (int i = 0; i < 5; ++i) {
            const int f  = tid + 256 * i;
            const int p  = f / 80;                 // K pair-row within chunk
            const int c2 = f % 80;                 // column pair
            const float* g = B + (size_t)(kc + 2 * p) * NCOLS + c0 + 2 * c2;
            lo[i] = *(const float2*)g;             // row 2p   (coalesced)
            hi[i] = *(const float2*)(g + NCOLS);   // row 2p+1 (coalesced)
        }
        __syncthreads();                           // previous tile fully consumed
        #pragma unroll
        for (int i = 0; i < 5; ++i) {
            const int f  = tid + 256 * i;
            const int p  = f / 80;
            const int c2 = f % 80;
            float4 st;                             // interleave K pair per column
            st.x = lo[i].x; st.y = hi[i].x;        // col 2c2   : {B[2p], B[2p+1]}
            st.z = lo[i].y; st.w = hi[i].y;        // col 2c2+1 : {B[2p], B[2p+1]}
            *(float4*)&Bs[(p * BNP + 2 * c2) * 2] = st;
        }
        __syncthreads();                           // tile visible to all waves

        // fused column sum-of-squares from the LDS-resident tile
        if (tid < BN) {
            #pragma unroll
            for (int p = 0; p < BK / 2; ++p) {
                const float2 v = *(const float2*)&Bs[(p * BNP + tid) * 2];
                csq = fmaf(v.x, v.x, csq);
                csq = fmaf(v.y, v.y, csq);
            }
        }

        // WMMA over the chunk: K steps of 4
        #pragma unroll
        for (int kk = 0; kk < BK; kk += 4) {
            const int kg = kc + kk;
            const v2f a = *(const v2f*)(Arow + kg + 2 * lh);   // A 16x4 fragment
            const int pb = (kk >> 1) + lh;                     // B pair-row for this half
            #pragma unroll
            for (int j = 0; j < 10; ++j) {
                const v2f b = *(const v2f*)&Bs[(pb * BNP + 16 * j + ll) * 2];
                acc[j] = __builtin_amdgcn_wmma_f32_16x16x4_f32(
                    /*neg_a=*/false, a, /*neg_b=*/false, b,
                    /*c_mod=*/(short)0, acc[j],
                    /*reuse_a=*/false, /*reuse_b=*/false);
            }
        }
    }

    // column inverse norms
    if (tid < BN) colinv[tid] = 1.0f / fmaxf(sqrtf(csq), 1e-12f);
    __syncthreads();

    // per-row parameters for this wave's 16-row strip (C/D: VGPR v -> rows v, v+8)
    int   rlab[8];
    float rga[8], rgd[8];
    #pragma unroll
    for (int v = 0; v < 8; ++v) {
        const int rrow = m0 + v + 8 * lh;
        rlab[v] = label[rrow];
        rga[v]  = gang[rrow];
        rgd[v]  = gadd[rrow];
    }

    // epilogue: clip cosine, apply margin on the label column only, scale by S
    #pragma unroll
    for (int j = 0; j < 10; ++j) {
        const int   c   = c0 + 16 * j + ll;
        const float inv = colinv[16 * j + ll];
        #pragma unroll
        for (int v = 0; v < 8; ++v) {
            float cosv = acc[j][v] * inv;
            cosv = fminf(fmaxf(cosv, -1.0f + EPS_C), 1.0f - EPS_C);
            float o;
            if (c == rlab[v]) {
                float th = acosf(cosv) + rga[v];
                th = fminf(fmaxf(th, EPS_C), PI_C - EPS_C);
                o = (__cosf(th) - rgd[v]) * S_C;
            } else {
                o = cosv * S_C;   // cos(clip(acos(x))) == x for clipped cosine
            }
            out[(size_t)(m0 + v + 8 * lh) * NCOLS + c] = o;
        }
    }
}

// ---------------------------------------------------------------------------
// Kernel 3: append embeddings (second tuple output) after the scaled logits.
// ---------------------------------------------------------------------------
__global__ void adaface_copy_emb(const float* __restrict__ emb,
                                 float* __restrict__ dst) {
    const int i = blockIdx.x * blockDim.x + threadIdx.x;   // float4 index
    ((float4*)dst)[i] = ((const float4*)emb)[i];
}

extern "C" void kernel_launch(void* const* d_in, const int* in_sizes, int n_in,
                              void* d_out, int out_size, void* d_ws, size_t ws_size,
                              hipStream_t stream) {
    const float* embeddings = (const float*)d_in[0];   // (512, 512)
    const float* kernelM    = (const float*)d_in[1];   // (512, 100000)
    const float* norms      = (const float*)d_in[2];   // (512,)
    const int*   label      = (const int*)d_in[3];     // (512,)
    float* out = (float*)d_out;

    float* gang = (float*)d_ws;            // 512 floats
    float* gadd = gang + NROWS;            // 512 floats

    adaface_margin_prep<<<1, NROWS, 0, stream>>>(norms, gang, gadd);

    dim3 grid(NCOLS / BN, NROWS / BM);     // 625 x 4
    adaface_gemm<<<grid, 256, 0, stream>>>(embeddings, kernelM, label, gang, gadd, out);

    // tuple output #2: embeddings, flat after the (N, C) logits
    float* dst = out + (size_t)NROWS * NCOLS;
    const int nvec4 = (NROWS * KDIM) / 4;  // 65536 float4
    adaface_copy_emb<<<nvec4 / 256, 256, 0, stream>>>(embeddings, dst);
}